// MyMarginLoss_63831803953309
// MI455X (gfx1250) — compile-verified
//
#include <hip/hip_runtime.h>

// Margin ranking loss, B=128, M=256, C=1024.
// loss = sum_{b,m in mask_mentions} [ (sum_c mask_cand * max(0, 1.5 - (s_t - s_c)) - 1.5) / (C-1) ]
// (target slot always unmasked and contributes exactly MARGIN=1.5, so it is
//  folded out by subtracting MARGIN once per mention instead of a per-element
//  c != target test.)
//
// Memory-bound: ~268 MB one-pass stream (f32 source + i32 candidate mask)
// -> ~11.5 us floor at 23.3 TB/s. Strategy: coalesced b128 loads, 2-3 VALU
// ops/element, WMMA-based wave reduction, LDS block reduction, 1 atomic/block.

#define MARGIN_F 1.5f
#define C_DIM 1024
#define ITERS 8            // C / (32 lanes * 4 floats per lane)
#define WAVES_PER_BLOCK 8  // 256 threads / wave32

typedef __attribute__((ext_vector_type(2))) float v2f;
typedef __attribute__((ext_vector_type(8))) float v8f;

__global__ void ML_zero_out(float* out) { out[0] = 0.0f; }

__global__ __launch_bounds__(256) void MyMarginLoss_kernel(
    const float* __restrict__ source,          // [BM, C] f32
    const int*   __restrict__ target,          // [BM] i32
    const int*   __restrict__ mask_mentions,   // [BM] i32 (0/1)
    const int*   __restrict__ mask_candidates, // [BM, C] i32 (0/1)
    float* __restrict__ out)                   // [1] f32
{
    __shared__ float wave_part[WAVES_PER_BLOCK];

    const int lane = threadIdx.x & 31;
    const int wave = threadIdx.x >> 5;
    const int bm   = blockIdx.x * WAVES_PER_BLOCK + wave;  // one mention per wave

    const float* __restrict__ src_row = source          + (size_t)bm * C_DIM;
    const int*   __restrict__ msk_row = mask_candidates + (size_t)bm * C_DIM;

    // Uniform per wave: target score and hinge constant k = MARGIN - s_t.
    const int   t   = target[bm];
    const float s_t = src_row[t];
    const float k   = MARGIN_F - s_t;

    // Pre-touch the far half of this wave's row (global_prefetch_b8).
    __builtin_prefetch(src_row + (size_t)lane * 4 + C_DIM / 2, 0, 1);

    // Streaming pass: 8 iterations of float4 + int4 per lane, fully coalesced
    // (each wave covers 128 consecutive floats / ints per iteration).
    float acc0 = 0.f, acc1 = 0.f, acc2 = 0.f, acc3 = 0.f;
    int base = lane * 4;
#pragma unroll
    for (int i = 0; i < ITERS; ++i) {
        const float4 s = *(const float4*)(src_row + base);
        const int4   m = *(const int4*)(msk_row + base);
        const float h0 = fmaxf(0.f, s.x + k);
        const float h1 = fmaxf(0.f, s.y + k);
        const float h2 = fmaxf(0.f, s.z + k);
        const float h3 = fmaxf(0.f, s.w + k);
        acc0 += m.x ? h0 : 0.f;
        acc1 += m.y ? h1 : 0.f;
        acc2 += m.z ? h2 : 0.f;
        acc3 += m.w ? h3 : 0.f;
        base += 128;
    }
    const float p = (acc0 + acc1) + (acc2 + acc3);  // per-lane partial

    // Wave32 reduction via V_WMMA_F32_16X16X4_F32 with B = ones:
    //   A[r][0] = p_r (lanes 0-15), A[r][2] = p_{r+16} (lanes 16-31), else 0
    //   => D[r][n] = p_r + p_{r+16} for every column n.
    // Summing the 8 D VGPRs per lane yields the half-sums (rows 0-7 in lanes
    // 0-15, rows 8-15 in lanes 16-31); one shfl_xor(16) completes the sum.
    // EXEC is all ones here: full blocks, uniform control flow.
    v2f a;      a[0] = p;   a[1] = 0.f;
    v2f bones;  bones[0] = 1.f; bones[1] = 1.f;
    v8f cacc = {};
    cacc = __builtin_amdgcn_wmma_f32_16x16x4_f32(
        /*neg_a=*/false, a, /*neg_b=*/false, bones,
        /*c_mod=*/(short)0, cacc, /*reuse_a=*/false, /*reuse_b=*/false);

    float halfsum = ((cacc[0] + cacc[1]) + (cacc[2] + cacc[3])) +
                    ((cacc[4] + cacc[5]) + (cacc[6] + cacc[7]));
    const float tot = halfsum + __shfl_xor(halfsum, 16, 32);

    if (lane == 0) {
        // Remove the target slot's fixed MARGIN contribution; scale by 1/(C-1);
        // gate by mask_mentions.
        const float contrib =
            mask_mentions[bm] ? (tot - MARGIN_F) * (1.0f / (float)(C_DIM - 1)) : 0.0f;
        wave_part[wave] = contrib;
    }
    __syncthreads();

    if (threadIdx.x == 0) {
        float blocksum = 0.f;
#pragma unroll
        for (int w = 0; w < WAVES_PER_BLOCK; ++w) blocksum += wave_part[w];
        atomicAdd(out, blocksum);  // 4096 atomics total
    }
}

extern "C" void kernel_launch(void* const* d_in, const int* in_sizes, int n_in,
                              void* d_out, int out_size, void* d_ws, size_t ws_size,
                              hipStream_t stream) {
    (void)n_in; (void)out_size; (void)d_ws; (void)ws_size;
    const float* source          = (const float*)d_in[0];
    const int*   target          = (const int*)d_in[1];
    const int*   mask_mentions   = (const int*)d_in[2];
    const int*   mask_candidates = (const int*)d_in[3];
    float*       out             = (float*)d_out;

    const int BM = in_sizes[1];             // B*M = 32768 mentions
    const int grid = BM / WAVES_PER_BLOCK;  // 32768 % 8 == 0 -> 4096 full blocks

    ML_zero_out<<<1, 1, 0, stream>>>(out);
    MyMarginLoss_kernel<<<grid, 256, 0, stream>>>(
        source, target, mask_mentions, mask_candidates, out);
}